// CrissCrossAttention_6519760355925
// MI455X (gfx1250) — compile-verified
//
#include <hip/hip_runtime.h>
#include <hip/hip_bf16.h>
#include <stdint.h>

// ---------------- problem constants ----------------
#define Bn   8
#define Cn   64
#define Hn   128
#define Wn   128
#define CQn  8
#define HWn  (Hn * Wn)        // 16384
#define HP   (Hn + 4)         // padded LDS stride (132): avoids 64-bank conflicts
#define EPSV 1e-5f

// ---------------- WMMA f32 16x16x4 -----------------
typedef __attribute__((ext_vector_type(2))) float v2f;
typedef __attribute__((ext_vector_type(8))) float v8f;
typedef __attribute__((ext_vector_type(4))) unsigned int u32x4;
typedef __attribute__((ext_vector_type(8))) unsigned int u32x8;

__device__ inline v8f wmma_f32(v2f a, v2f b, v8f c) {
    // D = A(16x4) * B(4x16) + C(16x16), all f32, wave32
    return __builtin_amdgcn_wmma_f32_16x16x4_f32(
        /*neg_a=*/false, a, /*neg_b=*/false, b,
        /*c_mod=*/(short)0, c, /*reuse_a=*/false, /*reuse_b=*/false);
}

// =====================================================================
// Kernel 1: projection  q = Wq x + bq ; k = Wk x + bk ; v = Wv x + bv
// One wave handles one 16-wide tile of HW for one batch.
// Writes BOTH the natural [b][ch][h][w] layout (row attention reads it
// contiguously) and a transposed [b][w][ch][h] layout (column attention
// reads it contiguously / via TDM).
// =====================================================================
__global__ __launch_bounds__(256) void proj_kernel(
    const float* __restrict__ x,
    const float* __restrict__ wq, const float* __restrict__ bq,
    const float* __restrict__ wk, const float* __restrict__ bk,
    const float* __restrict__ wv, const float* __restrict__ bv,
    float* __restrict__ q,  float* __restrict__ k,  float* __restrict__ v,
    float* __restrict__ qT, float* __restrict__ kT, float* __restrict__ vT)
{
    const int lane = threadIdx.x & 31;
    const int wave = threadIdx.x >> 5;
    const int half = lane >> 4;
    const int r    = lane & 15;

    const int tile = blockIdx.x * 8 + wave;          // 8192 tiles total
    const int b    = tile / (HWn / 16);
    const int n0   = (tile % (HWn / 16)) * 16;
    const int hrow = n0 / Wn;                        // 16 divides 128
    const int w0   = n0 % Wn;
    const float* xb = x + (size_t)b * Cn * HWn;

    // per-lane A-row base pointers (branchless: hoists wq/wk select)
    const float* wrow0 = (r < 8) ? (wq + r * Cn) : (wk + (r - 8) * Cn);
    const float* wrowv[4];
    for (int t = 0; t < 4; ++t) wrowv[t] = wv + (16 * t + r) * Cn;

    v8f acc[5];
    for (int t = 0; t < 5; ++t)
        acc[t] = (v8f){0.f,0.f,0.f,0.f,0.f,0.f,0.f,0.f};

    for (int k0 = 0; k0 < Cn; k0 += 4) {
        const int kk = k0 + 2 * half;
        v2f bm;
        bm.x = xb[(size_t)kk       * HWn + n0 + r];
        bm.y = xb[(size_t)(kk + 1) * HWn + n0 + r];

        v2f am; am.x = wrow0[kk]; am.y = wrow0[kk + 1];
        acc[0] = wmma_f32(am, bm, acc[0]);

        for (int t = 0; t < 4; ++t) {
            v2f av; av.x = wrowv[t][kk]; av.y = wrowv[t][kk + 1];
            acc[1 + t] = wmma_f32(av, bm, acc[1 + t]);
        }
    }

    // store q/k (both layouts)
    for (int i = 0; i < 8; ++i) {
        const int m = i + 8 * half;
        const float val = acc[0][i];
        if (m < 8) {
            const float o = val + bq[m];
            q [((size_t)(b * CQn + m)) * HWn + n0 + r] = o;
            qT[((size_t)(b * Wn + w0 + r) * CQn + m) * Hn + hrow] = o;
        } else {
            const float o = val + bk[m - 8];
            k [((size_t)(b * CQn + m - 8)) * HWn + n0 + r] = o;
            kT[((size_t)(b * Wn + w0 + r) * CQn + (m - 8)) * Hn + hrow] = o;
        }
    }
    // store v (both layouts)
    for (int t = 0; t < 4; ++t)
        for (int i = 0; i < 8; ++i) {
            const int c = 16 * t + i + 8 * half;
            const float o = acc[1 + t][i] + bv[c];
            v [((size_t)(b * Cn + c)) * HWn + n0 + r] = o;
            vT[((size_t)(b * Wn + w0 + r) * Cn + c) * Hn + hrow] = o;
        }
}

// =====================================================================
// Kernel 2: column attention. One workgroup per (b, w).
// V column slice is DMA'd into LDS by the Tensor Data Mover, with the
// D# pad fields inserting 4 pad DWORDs per 128 (LDS stride 132).
// =====================================================================
__global__ __launch_bounds__(256) void col_attn_kernel(
    const float* __restrict__ qT, const float* __restrict__ kT,
    const float* __restrict__ vT,
    float* __restrict__ outH, float* __restrict__ mH, float* __restrict__ sH)
{
    __shared__ float qs[CQn * Hn];     // [c][h]
    __shared__ float ks[CQn * Hn];     // [c][g]
    __shared__ float vs[Cn * HP];      // [c][g] padded stride 132 (TDM pad)
    __shared__ float Ps[Hn * HP];      // [h][g] padded

    const int tid  = threadIdx.x;
    const int lane = tid & 31;
    const int wave = tid >> 5;
    const int half = lane >> 4;
    const int r    = lane & 15;

    const int b = blockIdx.x / Wn;
    const int w = blockIdx.x % Wn;

    const float* qTb = qT + ((size_t)b * Wn + w) * (CQn * Hn);
    const float* kTb = kT + ((size_t)b * Wn + w) * (CQn * Hn);
    const float* vTb = vT + ((size_t)b * Wn + w) * (Cn * Hn);

    // ---- TDM: DMA the 32KB contiguous V slice into padded LDS ----
    if (wave == 0) {
        const uint64_t ga = (uint64_t)(uintptr_t)vTb;
        const unsigned lds_off = (unsigned)(uintptr_t)(void*)vs; // low 32 bits = LDS offset
        u32x4 g0;
        g0[0] = 1u;                                     // count=1 valid descriptor
        g0[1] = lds_off;                                // lds_addr
        g0[2] = (unsigned)(ga & 0xFFFFFFFFu);           // global_addr[31:0]
        g0[3] = (unsigned)((ga >> 32) & 0x01FFFFFFu)    // global_addr[56:32]
              | (2u << 30);                             // type=2 ("image")
        u32x8 g1;
        g1[0] = (2u << 16)      // data_size = 4B
              | (1u << 20)      // pad_enable
              | (6u << 22)      // pad_interval: 128 DWORDs
              | (3u << 25);     // pad_amount:   4 DWORDs
        g1[1] = (8192u & 0xFFFFu) << 16;  // tensor_dim0[15:0] @ bits 63:48
        g1[2] = (8192u >> 16)             // tensor_dim0[31:16]
              | (1u << 16);               // tensor_dim1 = 1
        g1[3] = (8192u << 16);            // tile_dim0 = 8192 @ bits 127:112
        g1[4] = 0u;                       // tile_dim1 = tile_dim2 = 0
        g1[5] = 8192u;                    // tensor_dim0_stride[31:0]
        g1[6] = 0u;
        g1[7] = 0u;
        asm volatile("tensor_load_to_lds %0, %1" :: "s"(g0), "s"(g1) : "memory");
    }

    // meanwhile: q/k slices (4KB each, fully coalesced)
    for (int idx = tid; idx < CQn * Hn; idx += 256) {
        qs[idx] = qTb[idx];
        ks[idx] = kTb[idx];
    }

    if (wave == 0) __builtin_amdgcn_s_wait_tensorcnt(0);
    __syncthreads();

    // E = Q^T K : wave -> h-tile, loop over g-tiles. K = 8 (2 k-steps).
    const int h0 = wave * 16;
    for (int gt = 0; gt < 8; ++gt) {
        const int g0i = gt * 16;
        v8f acc = (v8f){0.f,0.f,0.f,0.f,0.f,0.f,0.f,0.f};
        for (int k0 = 0; k0 < CQn; k0 += 4) {
            const int kk = k0 + 2 * half;
            v2f a, bm;
            a.x  = qs[kk * Hn + h0 + r];   a.y  = qs[(kk + 1) * Hn + h0 + r];
            bm.x = ks[kk * Hn + g0i + r];  bm.y = ks[(kk + 1) * Hn + g0i + r];
            acc = wmma_f32(a, bm, acc);
        }
        for (int i = 0; i < 8; ++i) {
            const int h = h0 + i + 8 * half;
            const int g = g0i + r;
            Ps[h * HP + g] = (h == g) ? -__builtin_inff() : acc[i];
        }
    }
    __syncthreads();

    // per-row softmax stats, P <- exp(P - m)
    if (tid < Hn) {
        const int h = tid;
        float m = -__builtin_inff();
        for (int g = 0; g < Hn; ++g) m = fmaxf(m, Ps[h * HP + g]);
        float s = 0.f;
        for (int g = 0; g < Hn; ++g) {
            const float e = __expf(Ps[h * HP + g] - m);
            Ps[h * HP + g] = e;
            s += e;
        }
        mH[(size_t)b * HWn + h * Wn + w] = m;
        sH[(size_t)b * HWn + h * Wn + w] = s;
    }
    __syncthreads();

    // out'_H = P * V^T : M=h(128) x N=c(64), K=g(128)
    for (int ct = 0; ct < 4; ++ct) {
        const int c0 = ct * 16;
        v8f acc = (v8f){0.f,0.f,0.f,0.f,0.f,0.f,0.f,0.f};
        for (int k0 = 0; k0 < Hn; k0 += 4) {
            const int kk = k0 + 2 * half;
            v2f a, bm;
            a.x  = Ps[(h0 + r) * HP + kk];  a.y  = Ps[(h0 + r) * HP + kk + 1];
            bm.x = vs[(c0 + r) * HP + kk];  bm.y = vs[(c0 + r) * HP + kk + 1];
            acc = wmma_f32(a, bm, acc);
        }
        for (int i = 0; i < 8; ++i) {
            const int h = h0 + i + 8 * half;
            const int c = c0 + r;
            outH[(((size_t)b * Cn + c) * Hn + h) * Wn + w] = acc[i];
        }
    }
}

// =====================================================================
// Kernel 3: row attention. One workgroup per (b, h). Natural layout is
// already contiguous along w. No diagonal mask.
// =====================================================================
__global__ __launch_bounds__(256) void row_attn_kernel(
    const float* __restrict__ q, const float* __restrict__ k,
    const float* __restrict__ v,
    float* __restrict__ outW, float* __restrict__ mW, float* __restrict__ sW)
{
    __shared__ float qs[CQn * Wn];
    __shared__ float ks[CQn * Wn];
    __shared__ float vs[Cn * HP];
    __shared__ float Ps[Wn * HP];

    const int tid  = threadIdx.x;
    const int lane = tid & 31;
    const int wave = tid >> 5;
    const int half = lane >> 4;
    const int r    = lane & 15;

    const int b = blockIdx.x / Hn;
    const int h = blockIdx.x % Hn;

    for (int idx = tid; idx < CQn * Wn; idx += 256) {
        const int c = idx >> 7, ww = idx & 127;
        qs[idx] = q[(((size_t)b * CQn + c) * Hn + h) * Wn + ww];
        ks[idx] = k[(((size_t)b * CQn + c) * Hn + h) * Wn + ww];
    }
    for (int idx = tid; idx < Cn * Wn; idx += 256) {
        const int c = idx >> 7, ww = idx & 127;
        vs[c * HP + ww] = v[(((size_t)b * Cn + c) * Hn + h) * Wn + ww];
    }
    __syncthreads();

    // E_W[w][vv] = sum_c q[c][w] * k[c][vv]
    const int w0 = wave * 16;
    for (int vt = 0; vt < 8; ++vt) {
        const int v0 = vt * 16;
        v8f acc = (v8f){0.f,0.f,0.f,0.f,0.f,0.f,0.f,0.f};
        for (int k0 = 0; k0 < CQn; k0 += 4) {
            const int kk = k0 + 2 * half;
            v2f a, bm;
            a.x  = qs[kk * Wn + w0 + r];  a.y  = qs[(kk + 1) * Wn + w0 + r];
            bm.x = ks[kk * Wn + v0 + r];  bm.y = ks[(kk + 1) * Wn + v0 + r];
            acc = wmma_f32(a, bm, acc);
        }
        for (int i = 0; i < 8; ++i)
            Ps[(w0 + i + 8 * half) * HP + v0 + r] = acc[i];
    }
    __syncthreads();

    if (tid < Wn) {
        const int ww = tid;
        float m = -__builtin_inff();
        for (int g = 0; g < Wn; ++g) m = fmaxf(m, Ps[ww * HP + g]);
        float s = 0.f;
        for (int g = 0; g < Wn; ++g) {
            const float e = __expf(Ps[ww * HP + g] - m);
            Ps[ww * HP + g] = e;
            s += e;
        }
        mW[(size_t)b * HWn + h * Wn + ww] = m;
        sW[(size_t)b * HWn + h * Wn + ww] = s;
    }
    __syncthreads();

    // out'_W[c][w] = sum_v P[w][v] * V[c][v]
    for (int ct = 0; ct < 4; ++ct) {
        const int c0 = ct * 16;
        v8f acc = (v8f){0.f,0.f,0.f,0.f,0.f,0.f,0.f,0.f};
        for (int k0 = 0; k0 < Wn; k0 += 4) {
            const int kk = k0 + 2 * half;
            v2f a, bm;
            a.x  = Ps[(w0 + r) * HP + kk];  a.y  = Ps[(w0 + r) * HP + kk + 1];
            bm.x = vs[(c0 + r) * HP + kk];  bm.y = vs[(c0 + r) * HP + kk + 1];
            acc = wmma_f32(a, bm, acc);
        }
        for (int i = 0; i < 8; ++i) {
            const int ww = w0 + i + 8 * half;
            const int c  = c0 + r;
            outW[(((size_t)b * Cn + c) * Hn + h) * Wn + ww] = acc[i];
        }
    }
}

// =====================================================================
// Kernel 4: flash-combine the two softmax halves, then per-(b,c)
// mean/var over HW, normalize, gamma-scale, residual add.
// =====================================================================
__global__ __launch_bounds__(256) void combine_norm_kernel(
    const float* __restrict__ x, const float* __restrict__ gamma,
    float* __restrict__ out,
    float* __restrict__ outH,          // in: unnormalized col-out, becomes s
    const float* __restrict__ outW,
    const float* __restrict__ mH, const float* __restrict__ sH,
    const float* __restrict__ mW, const float* __restrict__ sW)
{
    __shared__ float red1[256];
    __shared__ float red2[256];

    const int tid = threadIdx.x;
    const int bc  = blockIdx.x;
    const int b   = bc / Cn;
    const size_t base  = (size_t)bc * HWn;
    const size_t pbase = (size_t)b * HWn;

    float sum = 0.f, ssq = 0.f;
    for (int j = tid; j < HWn; j += 256) {
        const float mh = mH[pbase + j];
        const float mw = mW[pbase + j];
        const float m  = fmaxf(mh, mw);
        const float eh = __expf(mh - m);
        const float ew = __expf(mw - m);
        const float Z  = sH[pbase + j] * eh + sW[pbase + j] * ew;
        const float sv = (outH[base + j] * eh + outW[base + j] * ew) / Z;
        outH[base + j] = sv;
        sum += sv;
        ssq += sv * sv;
    }
    red1[tid] = sum; red2[tid] = ssq;
    __syncthreads();
    for (int off = 128; off > 0; off >>= 1) {
        if (tid < off) { red1[tid] += red1[tid + off]; red2[tid] += red2[tid + off]; }
        __syncthreads();
    }
    const float mean = red1[0] * (1.0f / HWn);
    const float var  = red2[0] * (1.0f / HWn) - mean * mean;
    const float inv  = rsqrtf(var + EPSV);
    const float g    = gamma[0];

    for (int j = tid; j < HWn; j += 256)
        out[base + j] = g * (outH[base + j] - mean) * inv + x[base + j];
}

// =====================================================================
// Host launcher
// =====================================================================
extern "C" void kernel_launch(void* const* d_in, const int* in_sizes, int n_in,
                              void* d_out, int out_size, void* d_ws, size_t ws_size,
                              hipStream_t stream) {
    (void)in_sizes; (void)n_in; (void)out_size; (void)ws_size;

    const float* x     = (const float*)d_in[0];
    const float* wq    = (const float*)d_in[1];
    const float* bq    = (const float*)d_in[2];
    const float* wk    = (const float*)d_in[3];
    const float* bk    = (const float*)d_in[4];
    const float* wv    = (const float*)d_in[5];
    const float* bv    = (const float*)d_in[6];
    const float* gamma = (const float*)d_in[7];
    float* out = (float*)d_out;

    // workspace layout (floats)
    float* ws = (float*)d_ws;
    const size_t nq = (size_t)Bn * CQn * HWn;   // 1,048,576
    const size_t nv = (size_t)Bn * Cn * HWn;    // 8,388,608
    const size_t np = (size_t)Bn * HWn;         //   131,072
    float* q    = ws;
    float* k    = q + nq;
    float* v    = k + nq;
    float* qT   = v + nv;
    float* kT   = qT + nq;
    float* vT   = kT + nq;
    float* outH = vT + nv;
    float* outW = outH + nv;
    float* mH   = outW + nv;
    float* sH   = mH + np;
    float* mW   = sH + np;
    float* sW   = mW + np;

    proj_kernel<<<(Bn * (HWn / 16)) / 8, 256, 0, stream>>>(
        x, wq, bq, wk, bk, wv, bv, q, k, v, qT, kT, vT);

    col_attn_kernel<<<Bn * Wn, 256, 0, stream>>>(qT, kT, vT, outH, mH, sH);
    row_attn_kernel<<<Bn * Hn, 256, 0, stream>>>(q, k, v, outW, mW, sW);

    combine_norm_kernel<<<Bn * Cn, 256, 0, stream>>>(
        x, gamma, out, outH, outW, mH, sH, mW, sW);
}